// _MHAEdge_61718680043591
// MI455X (gfx1250) — compile-verified
//
#include <hip/hip_runtime.h>
#include <hip/hip_bf16.h>

// Problem constants (match the reference)
#define N_NODES 50000
#define N_EDGES 800000
#define DIM     64
#define NHEAD   8
#define DKHEAD  8

typedef __attribute__((ext_vector_type(2))) float v2f;
typedef __attribute__((ext_vector_type(8))) float v8f;

// D(16x16,f32) = A(16x4,f32) x B(4x16,f32) + C  — CDNA5 V_WMMA_F32_16X16X4_F32
// 8-arg pattern: (neg_a, A, neg_b, B, c_mod, C, reuse_a, reuse_b)
__device__ __forceinline__ v8f wmma_f32_4(v2f a, v2f b, v8f c) {
    return __builtin_amdgcn_wmma_f32_16x16x4_f32(false, a, false, b,
                                                 (short)0, c, false, false);
}

// ---------------------------------------------------------------------------
// Zero-fill helper (d_ws / d_out are poisoned by the harness)
// ---------------------------------------------------------------------------
__global__ void _zero_kernel(float* __restrict__ p, int n) {
    int i = blockIdx.x * blockDim.x + threadIdx.x;
    if (i < n) p[i] = 0.0f;
}

// ---------------------------------------------------------------------------
// Node projections: Q = h@Wq, K = h@Wk, V = h@Wv.
// One wave per (16-row strip, weight). Per wave: 4 col-tiles x 16 k-steps of
// v_wmma_f32_16x16x4_f32.
//
// Fragment layouts (wave32, per ISA 7.12.2):
//  A 16x4:  lane l holds row m=l&15, elements K = 2*(l>>4), 2*(l>>4)+1
//  B 4x16:  lane l holds col n=l&15, elements K = 2*(l>>4), 2*(l>>4)+1
//  D 16x16: lane l, vgpr r holds element (m = r + 8*(l>>4), n = l&15)
// ---------------------------------------------------------------------------
__global__ void _node_proj_kernel(const float* __restrict__ h,
                                  const float* __restrict__ Wq,
                                  const float* __restrict__ Wk,
                                  const float* __restrict__ Wv,
                                  float* __restrict__ Q,
                                  float* __restrict__ K,
                                  float* __restrict__ V) {
    const int strip = blockIdx.x;          // 16-node strip, N_NODES/16 strips
    const int which = blockIdx.y;          // 0=Q, 1=K, 2=V
    const float* __restrict__ W  = (which == 0) ? Wq : (which == 1) ? Wk : Wv;
    float*       __restrict__ Out = (which == 0) ? Q : (which == 1) ? K : V;

    const int lane  = threadIdx.x & 31;
    const int m     = lane & 15;           // A-row / B-col / D-col index
    const int khalf = (lane >> 4) << 1;    // 0 or 2

    v8f acc0 = {}, acc1 = {}, acc2 = {}, acc3 = {};
    const int arow = strip * 16 + m;

    #pragma unroll
    for (int ks = 0; ks < 16; ++ks) {
        const int k0 = ks * 4 + khalf;
        v2f a;
        a.x = h[arow * DIM + k0];
        a.y = h[arow * DIM + k0 + 1];
        v2f b0, b1, b2, b3;
        b0.x = W[k0 * DIM + m];            b0.y = W[(k0 + 1) * DIM + m];
        b1.x = W[k0 * DIM + 16 + m];       b1.y = W[(k0 + 1) * DIM + 16 + m];
        b2.x = W[k0 * DIM + 32 + m];       b2.y = W[(k0 + 1) * DIM + 32 + m];
        b3.x = W[k0 * DIM + 48 + m];       b3.y = W[(k0 + 1) * DIM + 48 + m];
        acc0 = wmma_f32_4(a, b0, acc0);
        acc1 = wmma_f32_4(a, b1, acc1);
        acc2 = wmma_f32_4(a, b2, acc2);
        acc3 = wmma_f32_4(a, b3, acc3);
    }

    const int mst = (lane >> 4) * 8;       // D M-base: 0 or 8
    #pragma unroll
    for (int r = 0; r < 8; ++r) {
        const int orow = strip * 16 + mst + r;
        Out[orow * DIM +      m] = acc0[r];
        Out[orow * DIM + 16 + m] = acc1[r];
        Out[orow * DIM + 32 + m] = acc2[r];
        Out[orow * DIM + 48 + m] = acc3[r];
    }
}

// ---------------------------------------------------------------------------
// Edge kernel: 256 threads (8 waves), 128 edges per block.
//  Phase 1 (per wave, WMMA): E_tile[16,64] = e[tile]@We  -> LDS  (fused; never
//           materialized to HBM, saving ~205 MB of traffic)
//  Phase 2: scores s[edge,head] = dot(K[src],Q[dst])/sqrt(DK) -> LDS (L2 gathers)
//  Phase 3: alpha = exp(s*(1+E)); atomic segment-sums of alpha (-> Z) and
//           V[src]*alpha (-> Msum, accumulated directly into d_out).
// ---------------------------------------------------------------------------
__global__ void _edge_kernel(const float* __restrict__ e,
                             const float* __restrict__ We,
                             const float* __restrict__ Q,
                             const float* __restrict__ K,
                             const float* __restrict__ V,
                             const int*   __restrict__ src,
                             const int*   __restrict__ dst,
                             float* __restrict__ Z,
                             float* __restrict__ Msum) {
    __shared__ float E_lds[128 * DIM];     // 32 KB
    __shared__ float S_lds[128 * NHEAD];   //  4 KB

    const int tid   = threadIdx.x;         // 0..255
    const int wave  = tid >> 5;            // 0..7
    const int lane  = tid & 31;
    const int ebase = blockIdx.x * 128;    // block edge base
    const int tbase = ebase + wave * 16;   // this wave's 16-edge tile

    // ---- Phase 1: WMMA  E_tile = e[tbase:tbase+16, :] @ We -----------------
    const int m     = lane & 15;
    const int khalf = (lane >> 4) << 1;
    const int erow  = tbase + m;

    v8f acc0 = {}, acc1 = {}, acc2 = {}, acc3 = {};
    #pragma unroll
    for (int ks = 0; ks < 16; ++ks) {
        const int k0 = ks * 4 + khalf;
        v2f a;
        a.x = e[(size_t)erow * DIM + k0];
        a.y = e[(size_t)erow * DIM + k0 + 1];
        v2f b0, b1, b2, b3;
        b0.x = We[k0 * DIM + m];           b0.y = We[(k0 + 1) * DIM + m];
        b1.x = We[k0 * DIM + 16 + m];      b1.y = We[(k0 + 1) * DIM + 16 + m];
        b2.x = We[k0 * DIM + 32 + m];      b2.y = We[(k0 + 1) * DIM + 32 + m];
        b3.x = We[k0 * DIM + 48 + m];      b3.y = We[(k0 + 1) * DIM + 48 + m];
        acc0 = wmma_f32_4(a, b0, acc0);
        acc1 = wmma_f32_4(a, b1, acc1);
        acc2 = wmma_f32_4(a, b2, acc2);
        acc3 = wmma_f32_4(a, b3, acc3);
    }
    const int mst = (lane >> 4) * 8;
    #pragma unroll
    for (int r = 0; r < 8; ++r) {
        const int le = wave * 16 + mst + r;  // local edge index 0..127
        E_lds[le * DIM +      m] = acc0[r];
        E_lds[le * DIM + 16 + m] = acc1[r];
        E_lds[le * DIM + 32 + m] = acc2[r];
        E_lds[le * DIM + 48 + m] = acc3[r];
    }
    __syncthreads();

    // ---- Phase 2: scores (128 edges x 8 heads = 1024; 4 per thread) --------
    const float inv_sqrt_dk = 0.35355339059327373f;  // 1/sqrt(8)
    #pragma unroll
    for (int i = 0; i < 4; ++i) {
        const int idx = tid + i * 256;
        const int le  = idx >> 3;
        const int hh  = idx & 7;
        const int ge  = ebase + le;
        const int sn  = src[ge];
        const int dn  = dst[ge];
        const float* __restrict__ kp = K + (size_t)sn * DIM + hh * DKHEAD;
        const float* __restrict__ qp = Q + (size_t)dn * DIM + hh * DKHEAD;
        float dotv = 0.0f;
        #pragma unroll
        for (int k = 0; k < DKHEAD; ++k) dotv += kp[k] * qp[k];
        S_lds[idx] = dotv * inv_sqrt_dk;
    }
    __syncthreads();

    // ---- Phase 3: alpha + atomic segment sums (8192 elems; 32 per thread) --
    #pragma unroll 4
    for (int i = 0; i < 32; ++i) {
        const int idx = tid + i * 256;     // consecutive tids -> consecutive cols
        const int le  = idx >> 6;
        const int col = idx & 63;
        const int hh  = col >> 3;
        const int ge  = ebase + le;
        const float sc = S_lds[le * NHEAD + hh];
        const float a  = __expf(sc * (1.0f + E_lds[le * DIM + col]));
        const int sn = src[ge];
        const int dn = dst[ge];
        const float v = V[(size_t)sn * DIM + col];
        atomicAdd(&Z[(size_t)dn * DIM + col], a);
        atomicAdd(&Msum[(size_t)dn * DIM + col], v * a);
    }
}

// ---------------------------------------------------------------------------
// h_out = Msum / max(Z, 1e-6)   (Msum already accumulated in d_out)
// ---------------------------------------------------------------------------
__global__ void _finalize_kernel(const float* __restrict__ Z,
                                 float* __restrict__ out, int n) {
    int i = blockIdx.x * blockDim.x + threadIdx.x;
    if (i < n) {
        float z = fmaxf(Z[i], 1e-6f);
        out[i] = out[i] / z;
    }
}

extern "C" void kernel_launch(void* const* d_in, const int* in_sizes, int n_in,
                              void* d_out, int out_size, void* d_ws, size_t ws_size,
                              hipStream_t stream) {
    (void)in_sizes; (void)n_in; (void)out_size; (void)ws_size;

    const float* h  = (const float*)d_in[0];
    const float* e  = (const float*)d_in[1];
    const float* Wq = (const float*)d_in[2];
    const float* Wk = (const float*)d_in[3];
    const float* Wv = (const float*)d_in[4];
    const float* We = (const float*)d_in[5];
    const int*  src = (const int*)d_in[6];
    const int*  dst = (const int*)d_in[7];
    float* out = (float*)d_out;

    const size_t nelem = (size_t)N_NODES * DIM;   // 3.2M floats
    float* Q = (float*)d_ws;                      // ws layout: Q|K|V|Z
    float* K = Q + nelem;
    float* V = K + nelem;
    float* Z = V + nelem;

    const int zgrid = (int)((nelem + 255) / 256);
    _zero_kernel<<<zgrid, 256, 0, stream>>>(Z, (int)nelem);
    _zero_kernel<<<zgrid, 256, 0, stream>>>(out, (int)nelem);

    // Node projections: 3125 strips x {Q,K,V}, one wave each
    _node_proj_kernel<<<dim3(N_NODES / 16, 3), 32, 0, stream>>>(
        h, Wq, Wk, Wv, Q, K, V);

    // Fused edge projection + attention + scatter: 6250 blocks x 128 edges
    _edge_kernel<<<N_EDGES / 128, 256, 0, stream>>>(
        e, We, Q, K, V, src, dst, Z, out);

    _finalize_kernel<<<zgrid, 256, 0, stream>>>(Z, out, (int)nelem);
}